// MultiAPPNP_76012331205029
// MI455X (gfx1250) — compile-verified
//
#include <hip/hip_runtime.h>
#include <hip/hip_bf16.h>

typedef __attribute__((ext_vector_type(2))) float v2f;
typedef __attribute__((ext_vector_type(8))) float v8f;

#define APPNP_ALPHA 0.1f

// ---------------------------------------------------------------------------
// Degree accumulation: one thread per edge, float atomics (L2-resident).
// ---------------------------------------------------------------------------
__global__ void deg_kernel(const int* __restrict__ src, const int* __restrict__ dst,
                           float* __restrict__ degO, float* __restrict__ degI, int E) {
    int e = blockIdx.x * blockDim.x + threadIdx.x;
    if (e < E) {
        atomicAdd(&degO[src[e]], 1.0f);
        atomicAdd(&degI[dst[e]], 1.0f);
    }
}

// In-place: deg -> (deg > 0 ? rsqrt(max(deg,1)) : 0)
__global__ void norm_kernel(float* __restrict__ d, int n) {
    int i = blockIdx.x * blockDim.x + threadIdx.x;
    if (i < n) {
        float v = d[i];
        d[i] = (v > 0.0f) ? rsqrtf(fmaxf(v, 1.0f)) : 0.0f;
    }
}

// ---------------------------------------------------------------------------
// APPNP edge stage: 16 threads per edge, each moves a float4 (contiguous 256B
// per edge row -> fully coalesced gather), scatter via global_atomic_add_f32.
// f (25.6MB) and agg (25.6MB) both fit in the 192MB L2.
// ---------------------------------------------------------------------------
__global__ void edge_kernel(const int* __restrict__ src, const int* __restrict__ dst,
                            const float* __restrict__ f, const float* __restrict__ srcn,
                            float* __restrict__ agg, int E) {
    int t = blockIdx.x * blockDim.x + threadIdx.x;
    if (t >= E * 16) return;
    int e = t >> 4;
    int p = t & 15;
    int s = src[e];
    int d = dst[e];
    float sn = srcn[s];
    const float4 v = *(const float4*)(f + (size_t)s * 64 + p * 4);
    float* a = agg + (size_t)d * 64 + p * 4;
    atomicAdd(a + 0, v.x * sn);
    atomicAdd(a + 1, v.y * sn);
    atomicAdd(a + 2, v.z * sn);
    atomicAdd(a + 3, v.w * sn);
}

// APPNP node stage: f = (1-a) * agg * dst_norm + a * feat0   (f updated in place)
__global__ void node_kernel(const float* __restrict__ agg, const float* __restrict__ feat0,
                            const float* __restrict__ dstn, float* __restrict__ f, int N) {
    int t = blockIdx.x * blockDim.x + threadIdx.x;
    if (t >= N * 16) return;
    int n = t >> 4;
    int p = t & 15;
    float dn = dstn[n];
    const float4 a4 = *(const float4*)(agg + (size_t)n * 64 + p * 4);
    const float4 f0 = *(const float4*)(feat0 + (size_t)n * 64 + p * 4);
    float4 r;
    r.x = 0.9f * a4.x * dn + APPNP_ALPHA * f0.x;
    r.y = 0.9f * a4.y * dn + APPNP_ALPHA * f0.y;
    r.z = 0.9f * a4.z * dn + APPNP_ALPHA * f0.z;
    r.w = 0.9f * a4.w * dn + APPNP_ALPHA * f0.w;
    *(float4*)(f + (size_t)n * 64 + p * 4) = r;
}

// ---------------------------------------------------------------------------
// WMMA GEMM: Y[m0:m0+16, :] = act( X[m0:m0+16, 0:KDIM] @ W[KDIM, wcols] + b )
//
//   - X tile staged via CDNA5 async copy: global_load_async_to_lds_b128
//     (one 16B chunk per lane, coalesced), completed with s_wait_asynccnt 0.
//   - W staged transposed into LDS (column-major, padded stride KDIM+2 so the
//     v2f fragment reads are 8B-aligned and near conflict-free; pad columns
//     are pre-zeroed -> the K loop has NO branches and NO address math beyond
//     a ds offset increment).
//   - inner loop: ds_load_b64 (A) + ds_load_b64 (B) + v_wmma_f32_16x16x4_f32.
//   - EXEC is all-1s at every WMMA (divergence only in staging / epilogue
//     guards, fully reconverged before the MMA loop).
//
// A frag (16x4 f32): lanes 0-15 hold {K=k,k+1} of row M=lane,
//                    lanes 16-31 hold {K=k+2,k+3} of row M=lane-16.
// B frag (4x16 f32): mirrored with N in place of M.
// C/D (16x16 f32):   VGPR j = row (j + 8*half), col = lane&15.
// ---------------------------------------------------------------------------
template <int KDIM, int NCPAD, bool RELU>
__global__ void __launch_bounds__(2 * NCPAD)
gemm_wmma(const float* __restrict__ X, const float* __restrict__ W,
          const float* __restrict__ bias, float* __restrict__ Y,
          int nrows, int wcols, int ldY) {
    constexpr int KP = KDIM + 2;            // padded LDS strides (even => 8B aligned)
    __shared__ float sX[16 * KP];
    __shared__ float sW[NCPAD * KP];

    const int tid = threadIdx.x;
    const int m0 = blockIdx.x * 16;

    // --- async-stage the 16 x KDIM activation tile, 16B chunks per lane ---
    constexpr int CPR = KDIM / 4;           // float4 chunks per row
    for (int c = tid; c < 16 * CPR; c += 2 * NCPAD) {
        int r = c / CPR;
        int cc4 = c - r * CPR;
        int row = m0 + r;
        float* ldst = sX + r * KP + cc4 * 4;
        if (row < nrows) {
            unsigned lds = (unsigned)(size_t)ldst;
            unsigned long long ga =
                (unsigned long long)(size_t)(X + (size_t)row * KDIM + cc4 * 4);
            asm volatile("global_load_async_to_lds_b128 %0, %1, off"
                         :: "v"(lds), "v"(ga) : "memory");
        } else {
            *(float4*)ldst = make_float4(0.0f, 0.0f, 0.0f, 0.0f);
        }
    }

    // --- stage W transposed: sW[col*KP + k] = W[k][col], pad cols = 0 ---
    for (int idx = tid; idx < NCPAD * KDIM; idx += 2 * NCPAD) {
        int k = idx / NCPAD;
        int col = idx - k * NCPAD;
        float w = (col < wcols) ? W[(size_t)k * wcols + col] : 0.0f;
        sW[col * KP + k] = w;
    }

    asm volatile("s_wait_asynccnt 0x0" ::: "memory");
    __syncthreads();

    const int lane = tid & 31;
    const int wv   = tid >> 5;
    const int half = lane >> 4;             // 0: K pair {k,k+1}, 1: {k+2,k+3}
    const int lm   = lane & 15;             // row (A) / col (B) within the tile
    const int col  = wv * 16 + lm;

    const float* pA = sX + lm * KP + 2 * half;
    const float* pB = sW + col * KP + 2 * half;

    v8f c = {};
#pragma unroll
    for (int k = 0; k < KDIM; k += 4) {
        v2f a = *(const v2f*)(pA + k);
        v2f b = *(const v2f*)(pB + k);
        c = __builtin_amdgcn_wmma_f32_16x16x4_f32(
                /*neg_a=*/false, a, /*neg_b=*/false, b,
                /*c_mod=*/(short)0, c, /*reuse_a=*/false, /*reuse_b=*/false);
    }

    const float bv = (col < wcols) ? bias[col] : 0.0f;
    const int rbase = m0 + (half ? 8 : 0);
#pragma unroll
    for (int j = 0; j < 8; ++j) {
        int row = rbase + j;
        float v = c[j] + bv;
        if (RELU) v = fmaxf(v, 0.0f);
        if (row < nrows && col < wcols) {
            Y[(size_t)row * ldY + col] = v;
        }
    }
}

// ---------------------------------------------------------------------------
// Host-side launch sequence (graph-capture safe: only async ops on `stream`).
// ---------------------------------------------------------------------------
extern "C" void kernel_launch(void* const* d_in, const int* in_sizes, int n_in,
                              void* d_out, int out_size, void* d_ws, size_t ws_size,
                              hipStream_t stream) {
    const float* input_feat = (const float*)d_in[0];  // [N,64]
    const int*   src        = (const int*)d_in[1];    // [E]
    const int*   dst        = (const int*)d_in[2];    // [E]
    const float* W1  = (const float*)d_in[3];         // [64,128]
    const float* b1  = (const float*)d_in[4];         // [128]
    const float* W2  = (const float*)d_in[5];         // [128,128]
    const float* b2  = (const float*)d_in[6];         // [128]
    const float* Wh1 = (const float*)d_in[7];         // [128,64]
    const float* bh1 = (const float*)d_in[8];         // [64]
    const float* Wh2 = (const float*)d_in[9];         // [64,40]
    const float* bh2 = (const float*)d_in[10];        // [40]

    const int N = in_sizes[0] / 64;
    const int E = in_sizes[1];

    // Workspace layout (floats):
    //   srcn[N] | dstn[N] | f[N*64] | agg[N*64] | h1[N*64]
    //   H1 (N x 128) aliases the contiguous agg..h1 region after propagation.
    //   t  (N x 64)  aliases f after GEMM1 has consumed it.
    float* ws   = (float*)d_ws;
    float* srcn = ws;
    float* dstn = ws + N;
    float* f    = ws + (size_t)2 * N;
    float* agg  = f + (size_t)N * 64;
    float* h1   = agg + (size_t)N * 64;
    float* H1   = agg;   // [N,128] contiguous
    float* t    = f;     // [N,64] reuse

    float* outp  = (float*)d_out;               // [N,40]
    float* hlast = outp + (size_t)N * 40;       // [N,128]

    // Degrees -> symmetric norms.
    hipMemsetAsync(srcn, 0, sizeof(float) * (size_t)2 * N, stream);
    deg_kernel<<<(E + 255) / 256, 256, 0, stream>>>(src, dst, srcn, dstn, E);
    norm_kernel<<<(2 * N + 255) / 256, 256, 0, stream>>>(srcn, 2 * N);

    // f <- input_feat
    hipMemcpyAsync(f, input_feat, sizeof(float) * (size_t)N * 64,
                   hipMemcpyDeviceToDevice, stream);

    const int edge_grid = (int)(((size_t)E * 16 + 255) / 256);
    const int node_grid = (int)(((size_t)N * 16 + 255) / 256);

    for (int g = 0; g < 2; ++g) {
        if (g == 1) {
            // feat0 for gnn2 is the gnn1 result.
            hipMemcpyAsync(h1, f, sizeof(float) * (size_t)N * 64,
                           hipMemcpyDeviceToDevice, stream);
        }
        const float* feat0 = (g == 0) ? input_feat : h1;
        for (int k = 0; k < 10; ++k) {
            hipMemsetAsync(agg, 0, sizeof(float) * (size_t)N * 64, stream);
            edge_kernel<<<edge_grid, 256, 0, stream>>>(src, dst, f, srcn, agg, E);
            node_kernel<<<node_grid, 256, 0, stream>>>(agg, feat0, dstn, f, N);
        }
    }

    // MLP trunk + head on matrix cores.
    const int mt = (N + 15) / 16;
    gemm_wmma<64, 128, true ><<<mt, 256, 0, stream>>>(f,     W1,  b1,  H1,    N, 128, 128);
    gemm_wmma<128, 128, false><<<mt, 256, 0, stream>>>(H1,    W2,  b2,  hlast, N, 128, 128);
    gemm_wmma<128, 64,  true ><<<mt, 128, 0, stream>>>(hlast, Wh1, bh1, t,     N, 64,  64);
    gemm_wmma<64,  48,  false><<<mt, 96,  0, stream>>>(t,     Wh2, bh2, outp,  N, 40,  40);
}